// PBLNN_86199993630875
// MI455X (gfx1250) — compile-verified
//
#include <hip/hip_runtime.h>
#include <math.h>

typedef __attribute__((ext_vector_type(2))) float v2f;
typedef __attribute__((ext_vector_type(8))) float v8f;

#define BATCH 256
#define DIM   128
#define MAXIT 500
#define TOLER 1e-3f

__device__ __forceinline__ float softplusf(float x) {
    return fmaxf(x, 0.f) + log1pf(expf(-fabsf(x)));
}
__device__ __forceinline__ float sigmoidf(float x) {
    return 1.f / (1.f + expf(-x));
}
__device__ __forceinline__ float clipz(float w) {
    return fminf(fmaxf(w, 0.f), 1e10f);
}

// ---------------------------------------------------------------------------
// Init: y1 = 1.0, done = 0
// ---------------------------------------------------------------------------
__global__ void init_kernel(float* __restrict__ y1, int* __restrict__ done) {
    int idx = blockIdx.x * blockDim.x + threadIdx.x;
    if (idx < BATCH * DIM) y1[idx] = 1.0f;
    if (idx == 0) *done = 0;
}

// ---------------------------------------------------------------------------
// Generic one-time precompute linear: out[b,n] = act(sum_k in[b,k]*W[n,k]+bias[n]) * scale[n]
// act: 0=identity, 1=softplus.  One thread per output element (one-time cost,
// ~67 MFLOP total across all precompute launches -> irrelevant).
// ---------------------------------------------------------------------------
__global__ void lin_kernel(const float* __restrict__ in, const float* __restrict__ W,
                           const float* __restrict__ bias, const float* __restrict__ scale,
                           float* __restrict__ out, int N, int K, int act) {
    int idx = blockIdx.x * blockDim.x + threadIdx.x;
    int b = idx / N, n = idx - b * N;
    if (b >= BATCH) return;
    const float* ip = in + b * K;
    const float* wp = W + n * K;
    float s = bias ? bias[n] : 0.f;
    for (int k = 0; k < K; ++k) s += ip[k] * wp[k];
    if (act == 1) s = softplusf(s);
    if (scale) s *= scale[n];
    out[idx] = s;
}

// ---------------------------------------------------------------------------
// Fused fwd+bwd gradient of the ICNN w.r.t. y1 for a 16-row batch slab.
// 256 threads = 8 waves; wave w owns output columns [16w,16w+16).
// Six 16x128x128 GEMMs via V_WMMA_F32_16X16X4_F32 (full f32 precision).
// Writes delta = y_target - grad to global.
// ---------------------------------------------------------------------------
__global__ __launch_bounds__(256)
void grad_kernel(const float* __restrict__ y1, const float* __restrict__ yT,
                 const float* __restrict__ a0, const float* __restrict__ c0,
                 const float* __restrict__ a1, const float* __restrict__ c1,
                 const float* __restrict__ zu, const float* __restrict__ zu2,
                 const float* __restrict__ ga2,
                 const float* __restrict__ wy0, const float* __restrict__ wy1,
                 const float* __restrict__ wz1, const float* __restrict__ wz2,
                 float* __restrict__ delta) {
    __shared__ float bufA0[16 * DIM];  // y1 .* a0
    __shared__ float bufA1[16 * DIM];  // y1 .* a1
    __shared__ float bufZ [16 * DIM];  // z0 .* zu
    __shared__ float bufS0[16 * DIM];  // sigmoid(pre0)
    __shared__ float bufD1[16 * DIM];  // dL/dpre1
    __shared__ float bufD0[16 * DIM];  // dL/dpre0

    const int tid   = threadIdx.x;
    const int lane  = tid & 31;
    const int rbase = blockIdx.x * 16;      // batch-row base
    const int nbase = (tid >> 5) * 16;      // column tile base (per wave)
    const int ml    = lane & 15;            // A-row index / B,C column index
    const int koff  = (lane >> 4) * 2;      // half-wave K offset for f32 fragments

    // prologue: stage elementwise products for forward A operands
    for (int i = tid; i < 16 * DIM; i += 256) {
        int m = i >> 7, c = i & 127;
        int g = (rbase + m) * DIM + c;
        float yv = y1[g];
        bufA0[i] = yv * a0[g];
        bufA1[i] = yv * a1[g];
    }
    __syncthreads();

    // ---- stage 1: pre0 = (y.*a0) @ wy0^T + c0 ----
    {
        v8f acc = {};
        for (int k = 0; k < DIM; k += 4) {
            v2f a, b;
            a.x = bufA0[ml * DIM + k + koff];
            a.y = bufA0[ml * DIM + k + koff + 1];
            b.x = wy0[(nbase + ml) * DIM + k + koff];       // NT: B[k][n] = W[n][k]
            b.y = wy0[(nbase + ml) * DIM + k + koff + 1];
            acc = __builtin_amdgcn_wmma_f32_16x16x4_f32(false, a, false, b,
                                                        (short)0, acc, false, false);
        }
#pragma unroll
        for (int r = 0; r < 8; ++r) {
            int m = (lane < 16) ? r : (r + 8);
            int g = (rbase + m) * DIM + (nbase + ml);
            float pre0 = acc[r] + c0[g];
            bufS0[m * DIM + nbase + ml] = sigmoidf(pre0);
            bufZ [m * DIM + nbase + ml] = softplusf(pre0) * zu[g];
        }
    }
    __syncthreads();

    // ---- stage 2: pre1 = (z0.*zu) @ clip(wz1)^T + (y.*a1) @ wy1^T + c1 ----
    {
        v8f acc = {};
        for (int k = 0; k < DIM; k += 4) {
            v2f a, b;
            a.x = bufZ[ml * DIM + k + koff];
            a.y = bufZ[ml * DIM + k + koff + 1];
            b.x = clipz(wz1[(nbase + ml) * DIM + k + koff]);
            b.y = clipz(wz1[(nbase + ml) * DIM + k + koff + 1]);
            acc = __builtin_amdgcn_wmma_f32_16x16x4_f32(false, a, false, b,
                                                        (short)0, acc, false, false);
            a.x = bufA1[ml * DIM + k + koff];
            a.y = bufA1[ml * DIM + k + koff + 1];
            b.x = wy1[(nbase + ml) * DIM + k + koff];
            b.y = wy1[(nbase + ml) * DIM + k + koff + 1];
            acc = __builtin_amdgcn_wmma_f32_16x16x4_f32(false, a, false, b,
                                                        (short)0, acc, false, false);
        }
#pragma unroll
        for (int r = 0; r < 8; ++r) {
            int m = (lane < 16) ? r : (r + 8);
            int row = rbase + m, n = nbase + ml;
            float pre1 = acc[r] + c1[row * DIM + n];
            // dpre1 = zu2[b] * clip(wz2)[n] * sigmoid(pre1)
            bufD1[m * DIM + n] = zu2[row] * clipz(wz2[n]) * sigmoidf(pre1);
        }
    }
    __syncthreads();

    // ---- stage 3: dpre0 = (dpre1 @ clip(wz1)) .* zu .* s0 ----
    {
        v8f acc = {};
        for (int k = 0; k < DIM; k += 4) {
            v2f a, b;
            a.x = bufD1[ml * DIM + k + koff];
            a.y = bufD1[ml * DIM + k + koff + 1];
            b.x = clipz(wz1[(k + koff)     * DIM + nbase + ml]);  // NN: B[k][n] = W[k][n]
            b.y = clipz(wz1[(k + koff + 1) * DIM + nbase + ml]);
            acc = __builtin_amdgcn_wmma_f32_16x16x4_f32(false, a, false, b,
                                                        (short)0, acc, false, false);
        }
#pragma unroll
        for (int r = 0; r < 8; ++r) {
            int m = (lane < 16) ? r : (r + 8);
            int row = rbase + m, n = nbase + ml;
            bufD0[m * DIM + n] = acc[r] * zu[row * DIM + n] * bufS0[m * DIM + n];
        }
    }
    __syncthreads();

    // ---- stage 4: g = (dpre1@wy1).*a1 + (dpre0@wy0).*a0 + ga2 + 0.5*y1 ----
    {
        v8f acc4 = {}, acc5 = {};
        for (int k = 0; k < DIM; k += 4) {
            v2f a, b;
            a.x = bufD1[ml * DIM + k + koff];
            a.y = bufD1[ml * DIM + k + koff + 1];
            b.x = wy1[(k + koff)     * DIM + nbase + ml];
            b.y = wy1[(k + koff + 1) * DIM + nbase + ml];
            acc4 = __builtin_amdgcn_wmma_f32_16x16x4_f32(false, a, false, b,
                                                         (short)0, acc4, false, false);
            a.x = bufD0[ml * DIM + k + koff];
            a.y = bufD0[ml * DIM + k + koff + 1];
            b.x = wy0[(k + koff)     * DIM + nbase + ml];
            b.y = wy0[(k + koff + 1) * DIM + nbase + ml];
            acc5 = __builtin_amdgcn_wmma_f32_16x16x4_f32(false, a, false, b,
                                                         (short)0, acc5, false, false);
        }
#pragma unroll
        for (int r = 0; r < 8; ++r) {
            int m = (lane < 16) ? r : (r + 8);
            int g = (rbase + m) * DIM + (nbase + ml);
            float grad = acc4[r] * a1[g] + acc5[r] * a0[g] + ga2[g] + 0.5f * y1[g];
            delta[g] = yT[g] - grad;   // step direction (y - g)
        }
    }
}

// ---------------------------------------------------------------------------
// Single-block finisher: conv = mean(||delta||_row) < TOL ; frozen y1 update
// using done BEFORE this iteration's conv; then done |= conv (reference order).
// ---------------------------------------------------------------------------
__global__ __launch_bounds__(1024)
void fin_kernel(float* __restrict__ y1, const float* __restrict__ delta,
                int* __restrict__ done, float scale) {
    __shared__ float rs[BATCH];
    __shared__ float red[BATCH];
    __shared__ int convs;
    const int tid = threadIdx.x;
    const int doneOld = *done;                 // read by all threads before any write

    if (tid < BATCH) rs[tid] = 0.f;
    __syncthreads();

    // row sum-of-squares: 4 threads / row, 32 cols each
    {
        int row = tid >> 2, part = tid & 3;
        float p = 0.f;
        const float* dp = delta + row * DIM + part * 32;
        for (int c = 0; c < 32; ++c) { float d = dp[c]; p += d * d; }
        atomicAdd(&rs[row], p);
    }
    __syncthreads();

    if (tid < BATCH) red[tid] = sqrtf(rs[tid]);
    __syncthreads();
    for (int s = BATCH / 2; s > 0; s >>= 1) {
        if (tid < s) red[tid] += red[tid + s];
        __syncthreads();
    }
    if (tid == 0) convs = (red[0] / (float)BATCH) < TOLER ? 1 : 0;
    __syncthreads();

    // frozen update (uses done from BEFORE this iteration's conv)
    if (!doneOld) {
        for (int i = tid; i < BATCH * DIM; i += 1024)
            y1[i] += scale * delta[i];
    }
    __syncthreads();
    if (tid == 0) *done = doneOld | convs;
}

// ---------------------------------------------------------------------------
// out[b] = mean_j( y1[b,j] + CONVEX*y[b,j] ),  CONVEX = 1
// ---------------------------------------------------------------------------
__global__ void out_kernel(const float* __restrict__ y1, const float* __restrict__ yT,
                           float* __restrict__ out) {
    int b = blockIdx.x * blockDim.x + threadIdx.x;
    if (b >= BATCH) return;
    float s = 0.f;
    const float* yp = y1 + b * DIM;
    const float* tp = yT + b * DIM;
    for (int j = 0; j < DIM; ++j) s += yp[j] + tp[j];
    out[b] = s / (float)DIM;
}

// ---------------------------------------------------------------------------
extern "C" void kernel_launch(void* const* d_in, const int* in_sizes, int n_in,
                              void* d_out, int out_size, void* d_ws, size_t ws_size,
                              hipStream_t stream) {
    const float* x      = (const float*)d_in[0];
    const float* y      = (const float*)d_in[1];
    const float* wuu0_w = (const float*)d_in[2];
    const float* wuu0_b = (const float*)d_in[3];
    const float* wyu0_w = (const float*)d_in[4];
    const float* wyu0_b = (const float*)d_in[5];
    const float* wy0    = (const float*)d_in[6];
    const float* wu0_w  = (const float*)d_in[7];
    const float* wu0_b  = (const float*)d_in[8];
    const float* wuu1_w = (const float*)d_in[9];
    const float* wuu1_b = (const float*)d_in[10];
    const float* wzu1_w = (const float*)d_in[11];
    const float* wzu1_b = (const float*)d_in[12];
    const float* wz1    = (const float*)d_in[13];
    const float* wyu1_w = (const float*)d_in[14];
    const float* wyu1_b = (const float*)d_in[15];
    const float* wy1    = (const float*)d_in[16];
    const float* wu1_w  = (const float*)d_in[17];
    const float* wu1_b  = (const float*)d_in[18];
    const float* wzu2_w = (const float*)d_in[19];
    const float* wzu2_b = (const float*)d_in[20];
    const float* wz2    = (const float*)d_in[21];
    const float* wyu2_w = (const float*)d_in[22];
    const float* wyu2_b = (const float*)d_in[23];
    const float* wy2    = (const float*)d_in[24];
    // d_in[25]/d_in[26] (wu2_w/wu2_b) are constant in y -> no gradient contribution

    float* ws    = (float*)d_ws;
    float* u0    = ws;
    float* u1    = ws +  32768;
    float* a0    = ws +  65536;
    float* c0    = ws +  98304;
    float* a1    = ws + 131072;
    float* c1    = ws + 163840;
    float* zu    = ws + 196608;
    float* ga2   = ws + 229376;
    float* zu2   = ws + 262144;   // [256]
    float* y1    = ws + 262400;
    float* delta = ws + 295168;
    int*   done  = (int*)(ws + 327936);

    init_kernel<<<(BATCH * DIM + 255) / 256, 256, 0, stream>>>(y1, done);

    // one-time loop-invariant precompute (stream-serialized dependencies)
    lin_kernel<<<128, 256, 0, stream>>>(x,  wuu0_w, wuu0_b, nullptr, u0,  DIM, DIM, 1);
    lin_kernel<<<128, 256, 0, stream>>>(x,  wyu0_w, wyu0_b, nullptr, a0,  DIM, DIM, 0);
    lin_kernel<<<128, 256, 0, stream>>>(x,  wu0_w,  wu0_b,  nullptr, c0,  DIM, DIM, 0);
    lin_kernel<<<128, 256, 0, stream>>>(u0, wuu1_w, wuu1_b, nullptr, u1,  DIM, DIM, 1);
    lin_kernel<<<128, 256, 0, stream>>>(u0, wzu1_w, wzu1_b, nullptr, zu,  DIM, DIM, 1);
    lin_kernel<<<128, 256, 0, stream>>>(u0, wyu1_w, wyu1_b, nullptr, a1,  DIM, DIM, 0);
    lin_kernel<<<128, 256, 0, stream>>>(u0, wu1_w,  wu1_b,  nullptr, c1,  DIM, DIM, 0);
    lin_kernel<<<1,   256, 0, stream>>>(u1, wzu2_w, wzu2_b, nullptr, zu2, 1,   DIM, 1);
    lin_kernel<<<128, 256, 0, stream>>>(u1, wyu2_w, wyu2_b, wy2,     ga2, DIM, DIM, 0);

    // 500 sequential proximal iterations
    for (int i = 0; i < MAXIT; ++i) {
        grad_kernel<<<16, 256, 0, stream>>>(y1, y, a0, c0, a1, c1, zu, zu2, ga2,
                                            wy0, wy1, wz1, wz2, delta);
        float scale = 4.0f / (float)(i + 1);   // step/(i+1), step = 2*SMOOTH = 4
        fin_kernel<<<1, 1024, 0, stream>>>(y1, delta, done, scale);
    }

    out_kernel<<<1, 256, 0, stream>>>(y1, y, (float*)d_out);
}